// Discretizer_6554120094128
// MI455X (gfx1250) — compile-verified
//
#include <hip/hip_runtime.h>
#include <float.h>

typedef __attribute__((ext_vector_type(2))) float v2f;
typedef __attribute__((ext_vector_type(8))) float v8f;

#define DIM 64
#define CHUNK 128            // codes staged in LDS per outer iteration
#define PAD_STRIDE 66        // floats per LDS row (64 + 2 pad -> conflict-free A reads)
#define WG_THREADS 256
#define WAVES_PER_WG 8
#define TOKENS_PER_WAVE 16
#define TOKENS_PER_WG (WAVES_PER_WG * TOKENS_PER_WAVE)

// ---- kernel 1: per-code squared norms into workspace -----------------------
__global__ void vq_esq_kernel(const float* __restrict__ emb,
                              float* __restrict__ esq, int V) {
    int c = blockIdx.x * blockDim.x + threadIdx.x;
    if (c < V) {
        const float* row = emb + (size_t)c * DIM;
        float s = 0.f;
        #pragma unroll
        for (int i = 0; i < DIM; ++i) s += row[i] * row[i];
        esq[c] = s;
    }
}

// ---- kernel 2: fused WMMA-GEMM + argmin + gather ---------------------------
// D[M=code, N=token] = E_tile(16xK) x Z_tile^T(Kx16), accumulated over K=64
// via 16 chained V_WMMA_F32_16X16X4_F32. Argmin key = ||E||^2 - 2*cross.
__global__ void __launch_bounds__(WG_THREADS)
vq_argmin_kernel(const float* __restrict__ z, const float* __restrict__ emb,
                 const float* __restrict__ esq,
                 float* __restrict__ out_w, float* __restrict__ out_emb,
                 int NT, int V) {
    __shared__ float lds_tile[CHUNK * PAD_STRIDE];
    __shared__ float lds_esq[CHUNK];

    const int tid  = threadIdx.x;
    const int wave = tid >> 5;
    const int lane = tid & 31;
    const int lrow = lane & 15;   // token slot (B/N) == code row slot (A/M)
    const int hi   = lane >> 4;   // 0: K{0,1}, D-rows 0-7 ; 1: K{2,3}, D-rows 8-15

    const int wv_token0 = blockIdx.x * TOKENS_PER_WG + wave * TOKENS_PER_WAVE;
    const int my_token  = wv_token0 + lrow;

    // This wave's 16x64 z tile lives in registers as the B operands for the
    // 16 K-steps: B layout (4x16 f32): lanes 0-15 hold K0/K1, lanes 16-31 K2/K3.
    v2f zreg[16];
    {
        const float* zrow = z + (size_t)my_token * DIM;
        #pragma unroll
        for (int k = 0; k < 16; ++k)
            zreg[k] = *(const v2f*)(zrow + 4 * k + 2 * hi);
    }

    float best = FLT_MAX;
    int bestidx = 0;

    const int nchunk = V / CHUNK;
    for (int ch = 0; ch < nchunk; ++ch) {
        const int code0 = ch * CHUNK;
        __syncthreads();
        {   // cooperative stage: 128 codes x 64 f32 -> LDS (coalesced global reads)
            const float* src = emb + (size_t)code0 * DIM;
            #pragma unroll
            for (int j = 0; j < (CHUNK * DIM) / WG_THREADS; ++j) {
                int idx = tid + j * WG_THREADS;
                lds_tile[(idx >> 6) * PAD_STRIDE + (idx & 63)] = src[idx];
            }
            if (tid < CHUNK) lds_esq[tid] = esq[code0 + tid];
        }
        __syncthreads();

        #pragma unroll 2
        for (int t8 = 0; t8 < CHUNK / 16; ++t8) {
            v8f acc = {};
            // A layout (16x4 f32): lanes 0-15 hold K0/K1, lanes 16-31 K2/K3,
            // M = lane%16 -> code row (t8*16 + lrow) of the staged tile.
            const float* arow = &lds_tile[(t8 * 16 + lrow) * PAD_STRIDE + 2 * hi];
            #pragma unroll
            for (int k = 0; k < 16; ++k) {
                v2f a = *(const v2f*)(arow + 4 * k);
                acc = __builtin_amdgcn_wmma_f32_16x16x4_f32(
                        false, a, false, zreg[k], (short)0, acc, false, false);
            }
            // Per-lane argmin over the 8 code rows this lane holds.
            #pragma unroll
            for (int r = 0; r < 8; ++r) {
                int  cm  = t8 * 16 + r + 8 * hi;
                float key = lds_esq[cm] - 2.0f * acc[r];
                int  gc  = code0 + cm;
                bool take = (key < best);
                bestidx = take ? gc  : bestidx;
                best    = take ? key : best;
            }
        }
    }

    // Lane t holds rows {.. +0..7}, lane t+16 rows {.. +8..15} of the SAME
    // token: one xor-16 combine finishes the argmin (tie -> lower index).
    {
        float obest = __shfl_xor(best, 16, 32);
        int   oidx  = __shfl_xor(bestidx, 16, 32);
        bool take = (obest < best) || (obest == best && oidx < bestidx);
        best    = take ? obest : best;
        bestidx = take ? oidx  : bestidx;
    }

    if (lane < 16) out_w[my_token] = (float)bestidx;

    // Gather w_emb = emb_table[w] for this wave's 16 tokens (64 f32 each,
    // 2 per lane, coalesced).
    for (int t = 0; t < 16; ++t) {
        int idx = __shfl(bestidx, t, 32);
        const float* e = emb + (size_t)idx * DIM;
        float* o = out_emb + (size_t)(wv_token0 + t) * DIM;
        o[lane]      = e[lane];
        o[lane + 32] = e[lane + 32];
    }
}

extern "C" void kernel_launch(void* const* d_in, const int* in_sizes, int n_in,
                              void* d_out, int out_size, void* d_ws, size_t ws_size,
                              hipStream_t stream) {
    const float* z   = (const float*)d_in[0];   // (32,1024,64) f32
    const float* emb = (const float*)d_in[1];   // (8192,64)    f32
    const int NT = in_sizes[0] / DIM;           // 32768 tokens
    const int V  = in_sizes[1] / DIM;           // 8192 codes

    float* esq     = (float*)d_ws;              // V floats of scratch
    float* out_w   = (float*)d_out;             // indices (as f32), NT elems
    float* out_emb = out_w + NT;                // NT*64 floats

    vq_esq_kernel<<<(V + 255) / 256, 256, 0, stream>>>(emb, esq, V);
    vq_argmin_kernel<<<NT / TOKENS_PER_WG, WG_THREADS, 0, stream>>>(
        z, emb, esq, out_w, out_emb, NT, V);
}